// BDC_5265629905349
// MI455X (gfx1250) — compile-verified
//
#include <hip/hip_runtime.h>

// Problem constants (match reference)
#define B_   32
#define D_   640
#define M_   1024
#define EPS_ 1e-5f

#define TILE  128          // output tile edge per workgroup
#define NT    (D_ / TILE)  // 5 tiles per dimension
#define KSLAB 16           // K staged per LDS slab
#define LDP   20           // padded LDS row stride (floats): conflict-free, 16B-aligned rows
#define NS    (M_ / KSLAB) // 64 K-slabs

typedef __attribute__((ext_vector_type(2))) float v2f;
typedef __attribute__((ext_vector_type(8))) float v8f;
typedef int v4i_g __attribute__((vector_size(16)));   // matches builtin param type

#define AS1 __attribute__((address_space(1)))
#define AS3 __attribute__((address_space(3)))

#if __has_builtin(__builtin_amdgcn_global_load_async_to_lds_b128)
#define HAVE_ASYNC_COPY 1
#else
#define HAVE_ASYNC_COPY 0
#endif

#if __has_builtin(__builtin_amdgcn_s_wait_asynccnt)
#define WAIT_ASYNC0() __builtin_amdgcn_s_wait_asynccnt(0)
#else
#define WAIT_ASYNC0() asm volatile("s_wait_asynccnt 0x0" ::: "memory")
#endif

#if HAVE_ASYNC_COPY
// 16B global -> LDS async copy (tracked by ASYNCcnt, no VGPR round-trip)
__device__ __forceinline__ void async_cp16(const float* g, float* l) {
    __builtin_amdgcn_global_load_async_to_lds_b128(
        (AS1 v4i_g*)g, (AS3 v4i_g*)l, 0, 0);
}
#endif

// ---------------------------------------------------------------------------
// Kernel 0: per-row squared norms  sq[b*D + i] = sum_m x[b,i,m]^2
// ---------------------------------------------------------------------------
__global__ __launch_bounds__(256) void k_sqnorm(const float* __restrict__ x,
                                                float* __restrict__ sq) {
    const int wave = threadIdx.x >> 5;
    const int lane = threadIdx.x & 31;
    const int row  = blockIdx.x * 8 + wave;          // [0, B*D)
    const float* p = x + (size_t)row * M_;
    float s = 0.0f;
    for (int t = lane; t < M_; t += 32) {
        const float v = p[t];
        s = fmaf(v, v, s);
    }
    for (int off = 16; off > 0; off >>= 1) s += __shfl_xor(s, off, 32);
    if (lane == 0) sq[row] = s;
}

// ---------------------------------------------------------------------------
// Kernel 1: symmetric Gram via f32 WMMA (V_WMMA_F32_16X16X4_F32), fused dcov.
//   Block = one (ti <= tj) 128x128 tile pair of one batch; 8 wave32s,
//   each wave owns a 32x64 sub-tile = 2x4 16x16 f32 accumulators.
//   K staged through LDS in slabs of 16, double-buffered with
//   GLOBAL_LOAD_ASYNC_TO_LDS_B128 when available.
// ---------------------------------------------------------------------------
__global__ __launch_bounds__(256) void k_gram_dcov(const float* __restrict__ x,
                                                   const float* __restrict__ sq,
                                                   float* __restrict__ dcov) {
    __shared__ __align__(16) float As[2][TILE * LDP];
    __shared__ __align__(16) float Bs[2][TILE * LDP];

    const int b = blockIdx.y;

    // map linear pairId -> (ti, tj) with ti <= tj
    int ti = 0, rem = blockIdx.x;
    while (rem >= NT - ti) { rem -= NT - ti; ++ti; }
    const int tj = ti + rem;

    const int rowBase = ti * TILE;
    const int colBase = tj * TILE;
    const float* xb = x + (size_t)b * D_ * M_;

    const int tid  = threadIdx.x;
    const int wave = tid >> 5;
    const int lane = tid & 31;
    const int wm = wave & 3;   // 4 strips of 32 rows
    const int wn = wave >> 2;  // 2 strips of 64 cols
    const int lm = lane & 15;  // M (for A) / N (for B) within 16x16 tile
    const int kh = lane >> 4;  // which k-pair this lane holds

    // Per-thread staging assignment: elements e0 = tid, e1 = tid+256 of the
    // 512 float4s per 128x16 slab. Hoisted out of the K loop.
    const int r0 = tid >> 2,          kq0 = tid & 3;
    const int r1 = (tid + 256) >> 2,  kq1 = (tid + 256) & 3;
    const float* gA0 = xb + (size_t)(rowBase + r0) * M_ + kq0 * 4;
    const float* gA1 = xb + (size_t)(rowBase + r1) * M_ + kq1 * 4;
    const float* gB0 = xb + (size_t)(colBase + r0) * M_ + kq0 * 4;
    const float* gB1 = xb + (size_t)(colBase + r1) * M_ + kq1 * 4;
    const int lo0 = r0 * LDP + kq0 * 4;
    const int lo1 = r1 * LDP + kq1 * 4;

    v8f c[2][4] = {};

    // One slab of WMMA work: 4 k-steps x (2 A-frags, 4 B-frags, 8 WMMAs)
    auto compute_slab = [&](const float* Ac, const float* Bc) {
#pragma unroll
        for (int kk = 0; kk < 4; ++kk) {
            const int kl = kk * 4 + kh * 2;
            v2f a[2], bf[4];
#pragma unroll
            for (int mt = 0; mt < 2; ++mt)
                a[mt] = *(const v2f*)(&Ac[(wm * 32 + mt * 16 + lm) * LDP + kl]);
#pragma unroll
            for (int nt = 0; nt < 4; ++nt)
                bf[nt] = *(const v2f*)(&Bc[(wn * 64 + nt * 16 + lm) * LDP + kl]);
#pragma unroll
            for (int mt = 0; mt < 2; ++mt)
#pragma unroll
                for (int nt = 0; nt < 4; ++nt)
                    c[mt][nt] = __builtin_amdgcn_wmma_f32_16x16x4_f32(
                        false, a[mt], false, bf[nt], (short)0, c[mt][nt],
                        false, false);
        }
    };

#if HAVE_ASYNC_COPY
    // ---- async double-buffered pipeline ----
    // prologue: slab 0 -> buffer 0
    async_cp16(gA0, &As[0][lo0]);
    async_cp16(gA1, &As[0][lo1]);
    async_cp16(gB0, &Bs[0][lo0]);
    async_cp16(gB1, &Bs[0][lo1]);
    for (int s = 0; s < NS; ++s) {
        const int cur = s & 1;
        WAIT_ASYNC0();          // my slab-s copies have landed in LDS
        __syncthreads();        // everyone's copies landed; prior readers done
        if (s + 1 < NS) {       // prefetch slab s+1 into the other buffer
            const int nxt = cur ^ 1;
            const int ks = (s + 1) * KSLAB;
            async_cp16(gA0 + ks, &As[nxt][lo0]);
            async_cp16(gA1 + ks, &As[nxt][lo1]);
            async_cp16(gB0 + ks, &Bs[nxt][lo0]);
            async_cp16(gB1 + ks, &Bs[nxt][lo1]);
        }
        compute_slab(As[cur], Bs[cur]);
        // no trailing barrier: ds reads are drained before the next top barrier
    }
#else
    // ---- synchronous fallback: single buffer, unrolled staging ----
    for (int s = 0; s < NS; ++s) {
        const int ks = s * KSLAB;
        const float4 a0 = *(const float4*)(gA0 + ks);
        const float4 a1 = *(const float4*)(gA1 + ks);
        const float4 b0 = *(const float4*)(gB0 + ks);
        const float4 b1 = *(const float4*)(gB1 + ks);
        *(float4*)(&As[0][lo0]) = a0;
        *(float4*)(&As[0][lo1]) = a1;
        *(float4*)(&Bs[0][lo0]) = b0;
        *(float4*)(&Bs[0][lo1]) = b1;
        __syncthreads();
        compute_slab(As[0], Bs[0]);
        __syncthreads();
    }
#endif

    // Fused epilogue: dcov = sqrt(temp * max(d_i + d_j - 2*G, 0) + eps),
    // mirrored store for off-diagonal tile pairs (exact symmetry for free).
    const float temperature = 1.0f / (2.0f * (float)D_ * (float)M_);
    const float* sqb = sq + b * D_;
#pragma unroll
    for (int mt = 0; mt < 2; ++mt) {
#pragma unroll
        for (int nt = 0; nt < 4; ++nt) {
            const int gj = colBase + wn * 64 + nt * 16 + lm;  // N = lane%16
            const float sj = sqb[gj];
#pragma unroll
            for (int r = 0; r < 8; ++r) {
                const int gi = rowBase + wm * 32 + mt * 16 + r + 8 * kh;  // M = r + 8*(lane/16)
                const float g = c[mt][nt][r];
                float dv = sqb[gi] + sj - 2.0f * g;
                dv = fmaxf(dv, 0.0f);
                dv = sqrtf(fmaf(temperature, dv, EPS_));
                dcov[((size_t)b * D_ + gi) * D_ + gj] = dv;
                if (ti != tj) dcov[((size_t)b * D_ + gj) * D_ + gi] = dv;
            }
        }
    }
}

// ---------------------------------------------------------------------------
// Kernel 2: dcov row sums (dcov symmetric -> row sums == col sums)
// ---------------------------------------------------------------------------
__global__ __launch_bounds__(256) void k_rowsum(const float* __restrict__ dcov,
                                                float* __restrict__ rsum) {
    const int wave = threadIdx.x >> 5;
    const int lane = threadIdx.x & 31;
    const int row  = blockIdx.x * 8 + wave;          // [0, B*D)
    const float* p = dcov + (size_t)row * D_;
    float s = 0.0f;
    for (int t = lane; t < D_; t += 32) s += p[t];
    for (int off = 16; off > 0; off >>= 1) s += __shfl_xor(s, off, 32);
    if (lane == 0) rsum[row] = s;
}

// ---------------------------------------------------------------------------
// Kernel 3: total sum per batch
// ---------------------------------------------------------------------------
__global__ __launch_bounds__(256) void k_tot(const float* __restrict__ rsum,
                                             float* __restrict__ tot) {
    __shared__ float red[8];
    const int b    = blockIdx.x;
    const int tid  = threadIdx.x;
    const int wave = tid >> 5;
    const int lane = tid & 31;
    float s = 0.0f;
    for (int t = tid; t < D_; t += 256) s += rsum[b * D_ + t];
    for (int off = 16; off > 0; off >>= 1) s += __shfl_xor(s, off, 32);
    if (lane == 0) red[wave] = s;
    __syncthreads();
    if (wave == 0) {
        s = (lane < 8) ? red[lane] : 0.0f;
        for (int off = 4; off > 0; off >>= 1) s += __shfl_xor(s, off, 32);
        if (lane == 0) tot[b] = s;
    }
}

// ---------------------------------------------------------------------------
// Kernel 4: double-centering + upper-triangle gather.
//   p(i,j) = i*D - i*(i-1)/2 + (j - i)   (row-major triu enumeration)
// ---------------------------------------------------------------------------
__global__ __launch_bounds__(256) void k_gather(const float* __restrict__ dcov,
                                                const float* __restrict__ rsum,
                                                const float* __restrict__ tot,
                                                float* __restrict__ out) {
    const int i = blockIdx.x;   // 0..D-1
    const int b = blockIdx.y;   // 0..B-1
    const float invD  = 1.0f / (float)D_;
    const float tmean = tot[b] * invD * invD;
    const float rmi   = rsum[b * D_ + i] * invD;
    const size_t nTri = (size_t)D_ * (D_ + 1) / 2;    // 205120
    const size_t base = (size_t)i * D_ - ((size_t)i * (size_t)(i > 0 ? i - 1 : 0)) / 2 - (size_t)i;
    const float* drow = dcov + ((size_t)b * D_ + i) * D_;
    const float* rsb  = rsum + b * D_;
    for (int j = i + threadIdx.x; j < D_; j += 256) {
        const float v = drow[j] - rmi - rsb[j] * invD + tmean;
        out[b * nTri + base + j] = v;
    }
}

// ---------------------------------------------------------------------------
extern "C" void kernel_launch(void* const* d_in, const int* in_sizes, int n_in,
                              void* d_out, int out_size, void* d_ws, size_t ws_size,
                              hipStream_t stream) {
    (void)in_sizes; (void)n_in; (void)out_size; (void)ws_size;
    const float* x = (const float*)d_in[0];
    float* out = (float*)d_out;

    // Workspace layout (floats): dcov[B*D*D] | sq[B*D] | rsum[B*D] | tot[B]
    float* dcov = (float*)d_ws;
    float* sq   = dcov + (size_t)B_ * D_ * D_;
    float* rsum = sq + (size_t)B_ * D_;
    float* tot  = rsum + (size_t)B_ * D_;

    k_sqnorm<<<dim3(B_ * D_ / 8), dim3(256), 0, stream>>>(x, sq);

    dim3 g1(NT * (NT + 1) / 2, B_);                 // 15 tile pairs x 32 batches
    k_gram_dcov<<<g1, dim3(256), 0, stream>>>(x, sq, dcov);

    k_rowsum<<<dim3(B_ * D_ / 8), dim3(256), 0, stream>>>(dcov, rsum);
    k_tot<<<dim3(B_), dim3(256), 0, stream>>>(rsum, tot);

    dim3 g3(D_, B_);
    k_gather<<<g3, dim3(256), 0, stream>>>(dcov, rsum, tot, out);
}